// Attention_84988812853333
// MI455X (gfx1250) — compile-verified
//
#include <hip/hip_runtime.h>

#define NB   64      // sequences
#define NH   32      // query heads
#define NHKV 8       // kv heads
#define GQ   4       // q heads per kv head
#define DH   128     // head dim
#define BS   16      // tokens per cache block
#define BPB  64      // blocks per sequence
#define LMAX 1024
#define SCALE 0.08838834764831845f
#define PADL (LMAX + 4)   // +4 floats so the 4 P-rows land in distinct LDS banks

typedef __attribute__((ext_vector_type(2))) float v2f;
typedef __attribute__((ext_vector_type(8))) float v8f;

__device__ __forceinline__ v8f wmma4(v2f a, v2f b, v8f c) {
  // V_WMMA_F32_16X16X4_F32: D = A(16x4,f32) x B(4x16,f32) + C(16x16,f32)
  return __builtin_amdgcn_wmma_f32_16x16x4_f32(false, a, false, b,
                                               (short)0, c, false, false);
}

__global__ __launch_bounds__(128, 1) void paged_attn_decode(
    const float* __restrict__ q,   const float* __restrict__ kin,
    const float* __restrict__ vin, const float* __restrict__ kc,
    const float* __restrict__ vc,  const int* __restrict__ bt_all,
    const int* __restrict__ ctx_all, float* __restrict__ out)
{
  __shared__ __align__(16) float P[GQ][PADL];  // softmax rows, 4 real q heads
  const int b    = blockIdx.y;
  const int h    = blockIdx.x;                 // kv head
  const int tid  = threadIdx.x;
  const int wave = tid >> 5;
  const int lane = tid & 31;
  const int n16  = lane & 15;
  const int half = lane >> 4;
  const int ctx  = ctx_all[b];
  const int* bt  = bt_all + b * BPB;

  // ---------------- Q operand: 32 A-fragments (16x4 f32), K step = 4 dims.
  // f32 A 16x4 layout: row M = lane&15; elem e -> K = 2*half + e.
  // Fragment s covers d = 4*s .. 4*s+3, so lane holds Q[m][4s + 2h + e].
  v2f aq[32];
#pragma unroll
  for (int s = 0; s < 32; ++s) { aq[s][0] = 0.0f; aq[s][1] = 0.0f; }
  if (n16 < GQ) {
    const float* qp = q + ((size_t)b * NH + (h * GQ + n16)) * DH + 2 * half;
#pragma unroll
    for (int s = 0; s < 32; ++s) {
      const float2 t = *(const float2*)(qp + 4 * s);
      aq[s][0] = t.x;
      aq[s][1] = t.y;
    }
  }

  // ---------------- Phase 1: S = (Q K^T) * scale, one 16-token tile per iter.
  // f32 B 4x16 layout: column N = lane&15 (kv position), elem e -> K = 2h + e.
  // Tile i == cache block i of this sequence (BS == 16), so bt[i] is scalar.
  // Newest token (t == ctx-1) reads the fresh k input row.
  const int ntile = (ctx + 15) >> 4;
  for (int i = wave; i < ntile; i += 4) {
    const int t = i * 16 + n16;
    const float* kp;
    if (t == ctx - 1) {
      kp = kin + ((size_t)b * NHKV + h) * DH;
    } else {
      kp = kc + (((size_t)bt[i] * BS + n16) * NHKV + h) * DH;
    }
    kp += 2 * half;

    // Prefetch this wave's next tile (clamped, uniform control flow).
    {
      const int ipf = (i + 4 < ntile) ? (i + 4) : i;
      const float* kpf = kc + (((size_t)bt[ipf] * BS + n16) * NHKV + h) * DH;
      __builtin_prefetch(kpf + half * 16, 0, 0);
    }

    v8f s0 = {}, s1 = {};          // even/odd split halves the WMMA dep chain
#pragma unroll
    for (int s = 0; s < 32; ++s) {
      const float2 kv = *(const float2*)(kp + 4 * s);
      v2f bk;
      bk[0] = kv.x;
      bk[1] = kv.y;
      if (s & 1) s1 = wmma4(aq[s], bk, s1);
      else       s0 = wmma4(aq[s], bk, s0);
    }
    // C/D layout: VGPR r -> row r (half==0). Keep only the 4 real q rows.
    if (half == 0) {
#pragma unroll
      for (int r = 0; r < GQ; ++r)
        P[r][i * 16 + n16] = (s0[r] + s1[r]) * SCALE;
    }
  }
  __syncthreads();

  // ---------------- Phase 2: softmax, one wave per q row (4 waves, 4 rows).
  {
    const int r = wave;
    float mx = -3.0e38f;
    for (int t = lane; t < ctx; t += 32) mx = fmaxf(mx, P[r][t]);
#pragma unroll
    for (int off = 16; off > 0; off >>= 1) mx = fmaxf(mx, __shfl_xor(mx, off, 32));
    float sum = 0.0f;
    for (int t = lane; t < ctx; t += 32) {
      float e = __expf(P[r][t] - mx);
      P[r][t] = e;
      sum += e;
    }
#pragma unroll
    for (int off = 16; off > 0; off >>= 1) sum += __shfl_xor(sum, off, 32);
    const float inv = 1.0f / sum;
    const int lim = ntile * 16;          // zero tail: phase 3 needs no masking
    for (int t = lane; t < lim; t += 32) P[r][t] = (t < ctx) ? P[r][t] * inv : 0.0f;
  }
  __syncthreads();

  // ---------------- Phase 3: O = P V; each wave owns 32 output d columns.
  // Groups of 16 positions == one cache block -> bt[g] is wave-uniform.
  // A fragment (P, 16x4): elem e -> position 16g + 4j + 2h + e (from LDS).
  // B fragment (V, 4x16): column = d, elem e -> same positions (row 2h+e+4j).
  const int d0 = wave * 32;
  const float* vinrow = vin + ((size_t)b * NHKV + h) * DH;
  v8f o0a = {}, o0b = {}, o1a = {}, o1b = {};
  for (int g = 0; g < ntile; ++g) {
    const float* vbase = vc + ((size_t)bt[g] * BS * NHKV + h) * DH;

    // Prefetch the wave's 128B d-window of the next block.
    {
      const int gpf = (g + 1 < ntile) ? (g + 1) : g;
      __builtin_prefetch(
          vc + (((size_t)bt[gpf] * BS + n16) * NHKV + h) * DH + d0, 0, 0);
    }

#pragma unroll
    for (int j = 0; j < 4; ++j) {
      const int r0   = 4 * j + 2 * half;         // in-block row of elem 0
      const int pos0 = g * 16 + r0;
      const float* vp0 = (pos0     == ctx - 1) ? vinrow
                         : vbase + (size_t)r0 * (NHKV * DH);
      const float* vp1 = (pos0 + 1 == ctx - 1) ? vinrow
                         : vbase + (size_t)(r0 + 1) * (NHKV * DH);

      v2f pa;
      pa[0] = 0.0f; pa[1] = 0.0f;
      if (n16 < GQ) {
        const float2 pp = *(const float2*)&P[n16][pos0];
        pa[0] = pp.x; pa[1] = pp.y;
      }

      v2f b0, b1;
      b0[0] = vp0[d0 + n16];       b0[1] = vp1[d0 + n16];
      b1[0] = vp0[d0 + 16 + n16];  b1[1] = vp1[d0 + 16 + n16];

      if (j & 1) { o0b = wmma4(pa, b0, o0b); o1b = wmma4(pa, b1, o1b); }
      else       { o0a = wmma4(pa, b0, o0a); o1a = wmma4(pa, b1, o1a); }
    }
  }

  // D layout: VGPR r -> row r for half==0; column = lane&15 within the d tile.
  if (half == 0) {
#pragma unroll
    for (int r = 0; r < GQ; ++r) {
      float* op = out + ((size_t)b * NH + (h * GQ + r)) * DH;
      op[d0 + n16]      = o0a[r] + o0b[r];
      op[d0 + 16 + n16] = o1a[r] + o1b[r];
    }
  }
}

extern "C" void kernel_launch(void* const* d_in, const int* in_sizes, int n_in,
                              void* d_out, int out_size, void* d_ws, size_t ws_size,
                              hipStream_t stream) {
  const float* q   = (const float*)d_in[0];
  const float* k   = (const float*)d_in[1];
  const float* v   = (const float*)d_in[2];
  const float* kc  = (const float*)d_in[3];
  const float* vc  = (const float*)d_in[4];
  const int*   bt  = (const int*)d_in[5];
  const int*   ctx = (const int*)d_in[6];
  // d_in[7] (slot_mapping) is not needed: the newest token's K/V are taken
  // directly from the k/v inputs at position ctx-1 (caches stay immutable).
  (void)in_sizes; (void)n_in; (void)out_size; (void)d_ws; (void)ws_size;

  dim3 grid(NHKV, NB);
  dim3 block(128);
  hipLaunchKernelGGL(paged_attn_decode, grid, block, 0, stream,
                     q, k, v, kc, vc, bt, ctx, (float*)d_out);
}